// LogicLayer_89292370084190
// MI455X (gfx1250) — compile-verified
//
#include <hip/hip_runtime.h>
#include <stdint.h>

#define NUM_NEURONS 8192
#define INPUT_SIZE  8192
#define BATCH       2048
#define THREADS     256

typedef float floatx4 __attribute__((ext_vector_type(4)));

// ---------------------------------------------------------------------------
// Kernel 1: per-neuron softmax over 16 gate logits -> 4 affine coefficients.
// Every one of the 16 soft gates is affine in {1, a1, a2, a1*a2}, so
// out[b,n] = C0 + CA*a1 + CB*a2 + CP*a1*a2 with per-neuron constants.
// coef[n] = float4(CA, CB, CP, C0)
// ---------------------------------------------------------------------------
__global__ __launch_bounds__(THREADS) void coef_kernel(
    const float* __restrict__ w, float4* __restrict__ coef) {
  const int n = blockIdx.x * blockDim.x + threadIdx.x;
  if (n >= NUM_NEURONS) return;

  // per-gate coefficients of {a1, a2, p, 1}
  const float cA[16] = {0,0, 1,1, 0,0, 1, 1,-1,-1, 0, 0,-1,-1, 0,0};
  const float cB[16] = {0,0, 0,0, 1,1, 1, 1,-1,-1,-1,-1, 0, 0, 0,0};
  const float cP[16] = {0,1,-1,0,-1,0,-2,-1, 1, 2, 0, 1, 0, 1,-1,0};
  const float c1[16] = {0,0, 0,0, 0,0, 0, 0, 1, 1, 1, 1, 1, 1, 1,1};

  const float4* w4 = reinterpret_cast<const float4*>(w + (size_t)n * 16);
  float4 q0 = w4[0], q1 = w4[1], q2 = w4[2], q3 = w4[3];
  float v[16] = {q0.x,q0.y,q0.z,q0.w, q1.x,q1.y,q1.z,q1.w,
                 q2.x,q2.y,q2.z,q2.w, q3.x,q3.y,q3.z,q3.w};

  float m = v[0];
#pragma unroll
  for (int k = 1; k < 16; ++k) m = fmaxf(m, v[k]);

  float e[16], s = 0.f;
#pragma unroll
  for (int k = 0; k < 16; ++k) { e[k] = expf(v[k] - m); s += e[k]; }
  const float inv = 1.0f / s;

  float CA = 0.f, CB = 0.f, CP = 0.f, C0 = 0.f;
#pragma unroll
  for (int k = 0; k < 16; ++k) {
    const float p = e[k] * inv;
    CA = fmaf(p, cA[k], CA);
    CB = fmaf(p, cB[k], CB);
    CP = fmaf(p, cP[k], CP);
    C0 = fmaf(p, c1[k], C0);
  }
  coef[n] = make_float4(CA, CB, CP, C0);
}

// ---------------------------------------------------------------------------
// CDNA5 async data mover: global -> LDS, 16B per lane, ASYNCcnt-tracked.
// x is read exactly once per element -> non-temporal so the reused idx/coef
// tables keep L2/WGP$ residency.
// ---------------------------------------------------------------------------
__device__ __forceinline__ void async_copy_b128_nt(uint32_t lds_byte_off,
                                                   const void* gaddr) {
  asm volatile("global_load_async_to_lds_b128 %0, %1, off th:TH_LOAD_NT"
               :: "v"(lds_byte_off), "v"(gaddr)
               : "memory");
}
__device__ __forceinline__ void wait_asynccnt0() {
  asm volatile("s_wait_asynccnt 0x0" ::: "memory");
}

// ---------------------------------------------------------------------------
// Kernel 2: one block per batch row. Stage x[b,:] (32 KB) into LDS with async
// b128 copies, overlap the first neuron-quad's table loads with the staging
// wait, then gather-gather-FMA per neuron with float4 NT stores.
// ---------------------------------------------------------------------------
__global__ __launch_bounds__(THREADS) void logic_kernel(
    const float* __restrict__ x,
    const int4*  __restrict__ conn,   // (NUM_NEURONS,2) int32 -> int4 = 2 neurons
    const float4* __restrict__ coef,  // (CA,CB,CP,C0) per neuron
    float* __restrict__ out) {
  __shared__ float xrow[INPUT_SIZE];

  const int b = blockIdx.x;
  const int t = threadIdx.x;
  const float* xr = x + (size_t)b * INPUT_SIZE;

  // LDS byte offset of xrow (low 32 bits of the generic pointer are the LDS
  // address on AMDGPU).
  const uint32_t lds_base = (uint32_t)(uintptr_t)(&xrow[0]);

  // Stage the full row: 2048 float4 chunks, 8 per thread.
#pragma unroll
  for (int c = 0; c < INPUT_SIZE / (THREADS * 4); ++c) {
    const int e4 = c * THREADS + t;                 // float4 index in row
    async_copy_b128_nt(lds_base + (uint32_t)e4 * 16u,
                       (const void*)(xr + (size_t)e4 * 4));
  }

  // Prefetch iteration 0's tables while the async staging is in flight
  // (independent of LDS contents).
  const int nb0 = t * 4;
  int4   pi01 = conn[(nb0 >> 1) + 0];
  int4   pi23 = conn[(nb0 >> 1) + 1];
  float4 pk0  = coef[nb0 + 0];
  float4 pk1  = coef[nb0 + 1];
  float4 pk2  = coef[nb0 + 2];
  float4 pk3  = coef[nb0 + 3];

  wait_asynccnt0();      // this wave's async copies complete
  __syncthreads();       // all waves' copies visible

  // 8192 neurons: 4 per thread per iteration, 8 iterations.
#pragma unroll
  for (int j = 0; j < NUM_NEURONS / (THREADS * 4); ++j) {
    const int nb = j * (THREADS * 4) + t * 4;       // first neuron of this quad

    const int4   i01 = pi01;
    const int4   i23 = pi23;
    const float4 k0 = pk0, k1 = pk1, k2 = pk2, k3 = pk3;

    // Prefetch next iteration's tables (cache-resident after first row).
    if (j + 1 < NUM_NEURONS / (THREADS * 4)) {
      const int nbn = nb + THREADS * 4;
      pi01 = conn[(nbn >> 1) + 0];
      pi23 = conn[(nbn >> 1) + 1];
      pk0  = coef[nbn + 0];
      pk1  = coef[nbn + 1];
      pk2  = coef[nbn + 2];
      pk3  = coef[nbn + 3];
    }

    const float a0 = xrow[i01.x], b0 = xrow[i01.y];
    const float a1 = xrow[i01.z], b1 = xrow[i01.w];
    const float a2 = xrow[i23.x], b2 = xrow[i23.y];
    const float a3 = xrow[i23.z], b3 = xrow[i23.w];

    floatx4 r;
    r.x = fmaf(k0.z, a0 * b0, fmaf(k0.y, b0, fmaf(k0.x, a0, k0.w)));
    r.y = fmaf(k1.z, a1 * b1, fmaf(k1.y, b1, fmaf(k1.x, a1, k1.w)));
    r.z = fmaf(k2.z, a2 * b2, fmaf(k2.y, b2, fmaf(k2.x, a2, k2.w)));
    r.w = fmaf(k3.z, a3 * b3, fmaf(k3.y, b3, fmaf(k3.x, a3, k3.w)));

    // Streaming output: non-temporal 128-bit store.
    __builtin_nontemporal_store(
        r, reinterpret_cast<floatx4*>(out + (size_t)b * NUM_NEURONS + nb));
  }
}

// ---------------------------------------------------------------------------
extern "C" void kernel_launch(void* const* d_in, const int* in_sizes, int n_in,
                              void* d_out, int out_size, void* d_ws, size_t ws_size,
                              hipStream_t stream) {
  const float* x    = (const float*)d_in[0];
  const float* w    = (const float*)d_in[1];
  const int*   conn = (const int*)d_in[2];

  float4* coef = (float4*)d_ws;   // needs NUM_NEURONS * 16 B = 128 KB

  coef_kernel<<<NUM_NEURONS / THREADS, THREADS, 0, stream>>>(w, coef);
  logic_kernel<<<BATCH, THREADS, 0, stream>>>(
      x, (const int4*)conn, (const float4*)coef, (float*)d_out);
}